// truncated_krylov_layer_44641890075161
// MI455X (gfx1250) — compile-verified
//
#include <hip/hip_runtime.h>

// ---------------------------------------------------------------------------
// Truncated-Krylov graph layer for MI455X (gfx1250, wave32).
//   cat[n, b*128+f] = sum_e edge_vals[b,e] * x[edge_cols[b,e], f]  (scatter by rows)
//   out = cat @ W + bias          via V_WMMA_F32_16X16X4_F32 (fp32 tensor path)
// Memory-bound (~300 MB moved, 6.5 GFLOP) -> keep fp32 end-to-end.
// ---------------------------------------------------------------------------

#define N_NODES   50000
#define N_FEAT    128
#define N_BLOCKS  4
#define N_EDGES   800000
#define OUT_FEAT  128
#define CAT_FEAT  (N_FEAT * N_BLOCKS)   // 512

typedef __attribute__((ext_vector_type(2))) float v2f;
typedef __attribute__((ext_vector_type(8))) float v8f;

// ---------------------------------------------------------------------------
// Kernel 1: zero the concatenated-hops workspace (harness poisons d_ws).
// ---------------------------------------------------------------------------
__global__ void __launch_bounds__(256) tk_zero_ws(float4* __restrict__ ws, int n4) {
    int i = blockIdx.x * blockDim.x + threadIdx.x;
    if (i < n4) ws[i] = make_float4(0.f, 0.f, 0.f, 0.f);
}

// ---------------------------------------------------------------------------
// Kernel 2: SpMM scatter. One wave32 per edge; edge scalars are wave-uniform.
// Each lane handles 4 consecutive features (32 lanes x 4 = 128).
// Scatters directly into the hop-major concatenated layout [N, 4*128].
// ---------------------------------------------------------------------------
__global__ void __launch_bounds__(256) tk_spmm_scatter(
        const float* __restrict__ x,
        const int*   __restrict__ rows,
        const int*   __restrict__ cols,
        const float* __restrict__ vals,
        float*       __restrict__ cat) {
    const int lane = threadIdx.x & 31;
    const int waveInBlock = threadIdx.x >> 5;             // 8 waves per block
    const long eg = (long)blockIdx.x * 8 + waveInBlock;   // global edge id over all hops
    const long nE = (long)N_BLOCKS * N_EDGES;
    if (eg >= nE) return;

    const int b = (int)(eg / N_EDGES);                    // hop / block index
    const long idx = eg;                                  // flat [B, E] index
    const int r = rows[idx];
    const int c = cols[idx];
    const float v = vals[idx];

    const float4 xv = *(const float4*)(x + (long)c * N_FEAT + lane * 4);
    float* dst = cat + (long)r * CAT_FEAT + b * N_FEAT + lane * 4;

    atomicAdd(dst + 0, v * xv.x);
    atomicAdd(dst + 1, v * xv.y);
    atomicAdd(dst + 2, v * xv.z);
    atomicAdd(dst + 3, v * xv.w);
}

// ---------------------------------------------------------------------------
// Kernel 3: out[50000,128] = cat[50000,512] @ W[512,128] + bias.
// One block per 16-row tile (3125 blocks, exact). 8 waves; wave w owns output
// columns [16w, 16w+16). K swept in steps of 4 with V_WMMA_F32_16X16X4_F32.
//
// FP32 WMMA fragment layouts (ISA 7.12.2):
//   A 16x4 : lanes 0-15 -> M=lane,  VGPR{0,1} = K{0,1};  lanes 16-31 -> K{2,3}
//   B 4x16 : lanes 0-15 -> N=lane,  VGPR{0,1} = K{0,1};  lanes 16-31 -> K{2,3}
//   C 16x16: VGPR v: lanes 0-15 -> (M=v,   N=lane)
//                    lanes 16-31-> (M=v+8, N=lane-16)
// ---------------------------------------------------------------------------
__global__ void __launch_bounds__(256) tk_gemm_wmma(
        const float* __restrict__ cat,     // [N, 512]
        const float* __restrict__ W,       // [512, 128]
        const float* __restrict__ bias,    // [128]
        float*       __restrict__ out) {   // [N, 128]
    const int lane = threadIdx.x & 31;
    const int wave = threadIdx.x >> 5;     // 0..7 -> N tile
    const int l16  = lane & 15;
    const int half = lane >> 4;            // 0: K{0,1}, 1: K{2,3}

    const int m0 = blockIdx.x * 16;        // row tile base (50000 = 3125*16 exact)
    const int n0 = wave * 16;              // col tile base

    // Per-lane base pointers for the fragment elements this lane owns.
    const float* aPtr = cat + (long)(m0 + l16) * CAT_FEAT + half * 2;      // A[m=l16][k0+2*half + {0,1}]
    const float* bPtr = W + (long)(half * 2) * OUT_FEAT + (n0 + l16);      // W[k0+2*half + {0,1}][n]

    v8f c = {0.f, 0.f, 0.f, 0.f, 0.f, 0.f, 0.f, 0.f};

    #pragma unroll 4
    for (int k0 = 0; k0 < CAT_FEAT; k0 += 4) {
        // A: two consecutive floats -> single GLOBAL_LOAD_B64 (8B aligned).
        const float2 av = *(const float2*)(aPtr + k0);
        v2f a; a.x = av.x; a.y = av.y;

        // B: two row-strided floats of W (hot in L2: W is only 256 KB).
        v2f b;
        b.x = bPtr[(long)k0 * OUT_FEAT];
        b.y = bPtr[(long)k0 * OUT_FEAT + OUT_FEAT];

        // D = A x B + C   (8 args: neg_a, A, neg_b, B, c_mod, C, reuse_a, reuse_b)
        c = __builtin_amdgcn_wmma_f32_16x16x4_f32(
                false, a, false, b, (short)0, c, false, false);
    }

    // Epilogue: bias add + store per C layout.
    const float bv = bias[n0 + l16];
    float* o = out + (long)(m0 + half * 8) * OUT_FEAT + n0 + l16;
    #pragma unroll
    for (int v = 0; v < 8; ++v) {
        o[(long)v * OUT_FEAT] = c[v] + bv;
    }
}

// ---------------------------------------------------------------------------
// Host launcher. Inputs (setup_inputs order):
//   d_in[0] x            [50000,128] f32
//   d_in[1] edge_rows    [4,800000]  int32 (JAX default config downcasts int64)
//   d_in[2] edge_cols    [4,800000]  int32
//   d_in[3] edge_vals    [4,800000]  f32
//   d_in[4] shared_weight[512,128]   f32
//   d_in[5] output_bias  [128]       f32
// d_out: [50000,128] f32.  d_ws: cat buffer [50000,512] f32 (102.4 MB).
// ---------------------------------------------------------------------------
extern "C" void kernel_launch(void* const* d_in, const int* in_sizes, int n_in,
                              void* d_out, int out_size, void* d_ws, size_t ws_size,
                              hipStream_t stream) {
    const float* x    = (const float*)d_in[0];
    const int*   rows = (const int*)  d_in[1];
    const int*   cols = (const int*)  d_in[2];
    const float* vals = (const float*)d_in[3];
    const float* W    = (const float*)d_in[4];
    const float* bias = (const float*)d_in[5];
    float* out = (float*)d_out;
    float* cat = (float*)d_ws;

    // 1) zero cat workspace (required every call: ws is poisoned / reused)
    const int n4 = (N_NODES * CAT_FEAT) / 4;               // 6.4M float4
    tk_zero_ws<<<(n4 + 255) / 256, 256, 0, stream>>>((float4*)cat, n4);

    // 2) scatter SpMM over all hops (one wave per edge, 8 edges per block)
    const long nE = (long)N_BLOCKS * N_EDGES;              // 3.2M edges
    const int scatterBlocks = (int)((nE + 7) / 8);
    tk_spmm_scatter<<<scatterBlocks, 256, 0, stream>>>(x, rows, cols, vals, cat);

    // 3) dense GEMM with fp32 WMMA + bias epilogue
    tk_gemm_wmma<<<N_NODES / 16, 256, 0, stream>>>(cat, W, bias, out);
}